// Deconw_84628035600538
// MI455X (gfx1250) — compile-verified
//
#include <hip/hip_runtime.h>

typedef __attribute__((ext_vector_type(2))) float v2f;
typedef __attribute__((ext_vector_type(4))) float f4;
typedef __attribute__((ext_vector_type(8))) float v8f;

#define KH_ 4
#define KW_ 10
#define C_  64
#define H_  128
#define W_  512
#define B_  16
#define HO_ 131          // H + KH - 1
#define WO_ 521          // W + KW - 1
#define ROWS 19          // 16 + KH - 1
#define COLS 28          // 25 needed, padded to multiple of 4
#define CG   16          // channels staged per group (LDS budget)
#define CSTR (ROWS*COLS) // 532 floats per channel plane
#define WEXT 44          // zero-padded band row: index t = (j - n) + 15, t in [0,43]
#define WOFF 15          // offset so u = j - n = t - WOFF
#define WCH  (KH_*WEXT)  // 176 floats per channel of extended weights

__global__ __launch_bounds__(256) void deconw_wmma(
    const float* __restrict__ x, const float* __restrict__ w,
    const float* __restrict__ bias, float* __restrict__ out)
{
    __shared__ float sP[CG * CSTR];   // [cL][row*COLS+col]
    __shared__ float sWE[CG * WCH];   // [cL][kh][t], pre-flipped, zero-padded band row

    const int tid  = threadIdx.x;
    const int lane = tid & 31;
    const int wv   = tid >> 5;             // 8 waves per block
    const int ow0  = blockIdx.x * 16;
    const int oh0  = blockIdx.y * 16;
    const int bIdx = blockIdx.z;

    for (int cg = 0; cg < C_ / CG; ++cg) {
        const int cBase = cg * CG;

        // ---- stage input patch: float4 across channels -> coalesced 64B segments ----
        for (int s = tid; s < (CG * CSTR) / 4; s += 256) {
            int cq  = s & 3;                 // 4-channel quad within group
            int pos = s >> 2;                // row*COLS + col
            int row = pos / COLS;
            int col = pos - row * COLS;
            int gr  = oh0 + row - (KH_ - 1); // un-pad
            int gc  = ow0 + col - (KW_ - 1);
            f4 val = {0.f, 0.f, 0.f, 0.f};
            if (gr >= 0 && gr < H_ && gc >= 0 && gc < W_) {
                const float* src =
                    x + (((size_t)bIdx * H_ + gr) * W_ + gc) * C_ + cBase + cq * 4;
                val = *(const f4*)src;
            }
            int cL = cq * 4;
            float* dst = &sP[pos];
            dst[(size_t)(cL + 0) * CSTR] = val.x;
            dst[(size_t)(cL + 1) * CSTR] = val.y;
            dst[(size_t)(cL + 2) * CSTR] = val.z;
            dst[(size_t)(cL + 3) * CSTR] = val.w;
        }
        // ---- stage flipped weights as zero-padded extended band rows (tiny) ----
        for (int i = tid; i < CG * WCH; i += 256) {
            int cL = i / WCH;
            int r  = i - cL * WCH;
            int kh = r / WEXT;
            int u  = (r - kh * WEXT) - WOFF;      // u = j - n
            float v = 0.0f;
            if (u >= 0 && u < KW_)                // in-band: Wflip[kh][u]
                v = w[((KH_ - 1 - kh) * KW_ + (KW_ - 1 - u)) * C_ + cBase + cL];
            sWE[i] = v;
        }
        __syncthreads();

        // ---- compute: wave -> 2 channels; one 16x16 tile per channel = 28 WMMAs ----
        const int mn     = lane & 15;         // M (=oh) for A, N (=ow) for B
        const int kkBase = (lane >> 4) << 1;  // K sub-lane offset: 0 or 2

        for (int i = 0; i < 2; ++i) {
            const int cL = wv * 2 + i;
            const int c  = cBase + cL;
            const float bv = bias[c];
            v8f acc = {bv, bv, bv, bv, bv, bv, bv, bv};
            const float* pC = &sP[(size_t)cL * CSTR];
            const float* wC = &sWE[cL * WCH];
#pragma unroll
            for (int kh = 0; kh < KH_; ++kh) {
                // A: xp[oh0+mn+kh, ow0 + j0 + kk (+1)]  (kh-invariant base, j0 -> ds offset)
                const float* pRow = pC + (mn + kh) * COLS + kkBase;
                // B: wExt[kh][ (j0 + kk) - mn + WOFF ]  (unconditional, zero-padded)
                const float* wRow = wC + kh * WEXT + (WOFF + kkBase - mn);
#pragma unroll
                for (int jb = 0; jb < 7; ++jb) {  // K = 4*28 = 112
                    const int j0 = jb * 4;
                    v2f a;
                    a.x = pRow[j0];
                    a.y = pRow[j0 + 1];
                    v2f bm;
                    bm.x = wRow[j0];
                    bm.y = wRow[j0 + 1];
                    acc = __builtin_amdgcn_wmma_f32_16x16x4_f32(
                        false, a, false, bm, (short)0, acc, false, false);
                }
            }
            // ---- store D: VGPR v holds rows oh=v and oh=v+8; lanes = 16 contiguous ow ----
            const int owg = ow0 + mn;
            const int ohB = oh0 + ((lane >> 4) << 3);
            float* oBase = out + ((size_t)bIdx * C_ + c) * HO_ * (size_t)WO_;
#pragma unroll
            for (int v = 0; v < 8; ++v) {
                int oh = ohB + v;
                if (oh < HO_ && owg < WO_)
                    oBase[(size_t)oh * WO_ + owg] = acc[v];
            }
        }
        __syncthreads();
    }
}

extern "C" void kernel_launch(void* const* d_in, const int* in_sizes, int n_in,
                              void* d_out, int out_size, void* d_ws, size_t ws_size,
                              hipStream_t stream) {
    const float* x = (const float*)d_in[0];
    const float* w = (const float*)d_in[1];
    const float* b = (const float*)d_in[2];
    float* out = (float*)d_out;
    dim3 grid((WO_ + 15) / 16, (HO_ + 15) / 16, B_);
    deconw_wmma<<<grid, dim3(256), 0, stream>>>(x, w, b, out);
}